// RNNModel_11304353923592
// MI455X (gfx1250) — compile-verified
//
#include <hip/hip_runtime.h>
#include <cmath>

#define NN   2048
#define TT   8192
#define NBLK 64
#define ROWS 32        // rows of W_rec per block
#define NTH  256       // 8 wave32 per block
#define FPW  128       // WMMA fragments (K-chunks of 4) per wave: 512 K / 4

typedef float v2f __attribute__((ext_vector_type(2)));
typedef float v8f __attribute__((ext_vector_type(8)));

// Grid-wide barrier: monotonic counter, release/acquire at agent scope.
__device__ __forceinline__ void gridbar(unsigned* cnt, unsigned target) {
  __threadfence();                 // make this thread's global stores agent-visible
  __syncthreads();
  if (threadIdx.x == 0) {
    __hip_atomic_fetch_add(cnt, 1u, __ATOMIC_RELEASE, __HIP_MEMORY_SCOPE_AGENT);
    while (__hip_atomic_load(cnt, __ATOMIC_ACQUIRE, __HIP_MEMORY_SCOPE_AGENT) < target)
      __builtin_amdgcn_s_sleep(1);
  }
  __syncthreads();
  __threadfence();                 // acquire: invalidate stale near caches
}

__global__ __launch_bounds__(NTH, 1)
void rnn_persistent(const float* __restrict__ Iin,
                    const float* __restrict__ noise,
                    const float* __restrict__ Wrec,
                    const float* __restrict__ Win,
                    const float* __restrict__ Wout,
                    const float* __restrict__ bvec,
                    float* __restrict__ U,       // [T, N]
                    float* __restrict__ Y,       // [T, 2]
                    float* __restrict__ ubuf,    // ws: double-buffered u, 2*N floats
                    unsigned* __restrict__ cnt)  // ws: barrier counter (memset to 0)
{
  // 256 KB W slice (WMMA-A fragment layout) + 8 KB u + scratch: ~273 KB < 320 KB/WGP
  __shared__ __align__(16) float Wl[ROWS * NN];
  __shared__ __align__(16) float ul[NN];
  __shared__ float part[8][16];
  __shared__ float xl[ROWS];
  __shared__ float yr0[NTH];
  __shared__ float yr1[NTH];

  const int tid  = threadIdx.x;
  const int lane = tid & 31;
  const int wave = tid >> 5;
  const int g    = wave >> 2;      // row-group (0/1): 16 rows each
  const int q    = wave & 3;       // K quarter
  const int row0 = blockIdx.x * ROWS;

  // ---- one-time: load W_rec slice, swizzled into V_WMMA_F32_16X16X4 A layout ----
  // lanes 0-15: row m, K = {0,1}; lanes 16-31: row m, K = {2,3}  (ISA 7.12.2)
  for (int idx = tid; idx < ROWS * NN; idx += NTH) {
    const int r = idx >> 11;              // local row
    const int k = idx & (NN - 1);
    const float w = Wrec[(size_t)(row0 + r) * NN + k];
    const int gg = r >> 4, m = r & 15, c = k >> 2, kk = k & 3;
    Wl[(((gg * (NN / 4)) + c) * 32 + m + 16 * (kk >> 1)) * 2 + (kk & 1)] = w;
  }

  // per-row constants + state init (u_0 = tanh(0) = 0; ubuf zeroed by memset)
  float win0 = 0.f, win1 = 0.f, brow = 0.f;
  if (tid < ROWS) {
    win0 = Win[(row0 + tid) * 2 + 0];
    win1 = Win[(row0 + tid) * 2 + 1];
    brow = bvec[row0 + tid];
    xl[tid] = 0.f;
    U[row0 + tid] = 0.f;
  }
  float wo0[8], wo1[8];
  if (blockIdx.x == 0) {
#pragma unroll
    for (int j = 0; j < 8; ++j) {
      wo0[j] = Wout[tid + NTH * j];
      wo1[j] = Wout[NN + tid + NTH * j];
    }
  }
  __syncthreads();

  const v2f* Wl2  = (const v2f*)Wl;
  const int fbase = g * (NN / 4) + q * FPW;
  const int bofs  = (lane >> 4) << 1;    // 0 for lanes<16, 2 for lanes>=16

  for (unsigned t = 0; t < TT; ++t) {
    gridbar(cnt, NBLK * (t + 1));        // u_t globally visible in ubuf[t&1]

    // prefetch per-step scalars early (latency overlaps the matvec)
    float nz = 0.f, i0 = 0.f, i1 = 0.f;
    if (tid < ROWS) {
      nz = noise[(size_t)t * NN + row0 + tid];
      i0 = Iin[2 * t + 0];
      i1 = Iin[2 * t + 1];
    }

    // stage full u_t into LDS (8 KB)
    {
      const float4* src = (const float4*)(ubuf + (size_t)(t & 1) * NN);
      float4* dst = (float4*)ul;
      dst[tid]       = src[tid];
      dst[tid + NTH] = src[tid + NTH];
    }
    __syncthreads();

    // ---- WMMA matvec: this wave covers 16 rows x 512 K ----
    v8f a0 = {}, a1 = {}, a2 = {}, a3 = {};
#pragma unroll 2
    for (int c = 0; c < FPW; c += 4) {
      const int f  = fbase + c;
      const int ku = (q * FPW + c) * 4 + bofs;
      v2f A0 = Wl2[(f + 0) * 32 + lane];
      v2f A1 = Wl2[(f + 1) * 32 + lane];
      v2f A2 = Wl2[(f + 2) * 32 + lane];
      v2f A3 = Wl2[(f + 3) * 32 + lane];
      v2f B0 = *(const v2f*)&ul[ku + 0];   // B columns all = u chunk (broadcast)
      v2f B1 = *(const v2f*)&ul[ku + 4];
      v2f B2 = *(const v2f*)&ul[ku + 8];
      v2f B3 = *(const v2f*)&ul[ku + 12];
      a0 = __builtin_amdgcn_wmma_f32_16x16x4_f32(false, A0, false, B0, (short)0, a0, false, false);
      a1 = __builtin_amdgcn_wmma_f32_16x16x4_f32(false, A1, false, B1, (short)0, a1, false, false);
      a2 = __builtin_amdgcn_wmma_f32_16x16x4_f32(false, A2, false, B2, (short)0, a2, false, false);
      a3 = __builtin_amdgcn_wmma_f32_16x16x4_f32(false, A3, false, B3, (short)0, a3, false, false);
    }
    v8f acc = (a0 + a1) + (a2 + a3);

    // C layout: lanes 0-15 hold rows M=0..7 in v0..7 (any column), lanes 16-31 rows 8..15
    if (lane == 0) {
#pragma unroll
      for (int v = 0; v < 8; ++v) part[wave][v] = acc[v];
    } else if (lane == 16) {
#pragma unroll
      for (int v = 0; v < 8; ++v) part[wave][8 + v] = acc[v];
    }
    __syncthreads();

    // ---- row update: x_{t+1} = x_t + alpha*(-x_t + Wu + Win I + b + sigma n) ----
    if (tid < ROWS) {
      const int gg = tid >> 4, m = tid & 15;
      const float mv = (part[gg * 4 + 0][m] + part[gg * 4 + 1][m]) +
                       (part[gg * 4 + 2][m] + part[gg * 4 + 3][m]);
      const float rhs = mv + win0 * i0 + win1 * i1 + brow + 0.01f * nz;
      const float x   = xl[tid];
      const float xn  = x + 0.1f * (rhs - x);
      xl[tid] = xn;
      if (t + 1 < TT) {
        const float un = tanhf(xn);
        U[(size_t)(t + 1) * NN + row0 + tid] = un;
        ubuf[(size_t)((t + 1) & 1) * NN + row0 + tid] = un;
      }
    }

    // ---- readout y_t = W_out @ u_t : block 0, fixed-order tree (deterministic) ----
    if (blockIdx.x == 0) {
      float p0 = 0.f, p1 = 0.f;
#pragma unroll
      for (int j = 0; j < 8; ++j) {
        const float uv = ul[tid + NTH * j];
        p0 += wo0[j] * uv;
        p1 += wo1[j] * uv;
      }
      yr0[tid] = p0; yr1[tid] = p1;
      __syncthreads();
      for (int s = NTH / 2; s > 0; s >>= 1) {
        if (tid < s) { yr0[tid] += yr0[tid + s]; yr1[tid] += yr1[tid + s]; }
        __syncthreads();
      }
      if (tid == 0) { Y[2 * t] = yr0[0]; Y[2 * t + 1] = yr1[0]; }
    }
  }
}

extern "C" void kernel_launch(void* const* d_in, const int* in_sizes, int n_in,
                              void* d_out, int out_size, void* d_ws, size_t ws_size,
                              hipStream_t stream) {
  const float* I    = (const float*)d_in[0];   // [T, 2]
  const float* nz   = (const float*)d_in[1];   // [T, N]
  const float* Wrec = (const float*)d_in[2];   // [N, N]
  const float* Win  = (const float*)d_in[3];   // [N, 2]
  const float* Wout = (const float*)d_in[4];   // [2, N]
  const float* b    = (const float*)d_in[5];   // [N]
  float* U = (float*)d_out;                    // [T, N]
  float* Y = U + (size_t)TT * NN;              // [T, 2]

  unsigned* cnt = (unsigned*)d_ws;             // barrier counter @ offset 0
  float* ubuf   = (float*)((char*)d_ws + 256); // double-buffered u

  // zero barrier counter + u_0 (u_0 = tanh(0) = 0); capture-safe memset node
  hipMemsetAsync(d_ws, 0, 256 + 2 * NN * sizeof(float), stream);
  rnn_persistent<<<NBLK, NTH, 0, stream>>>(I, nz, Wrec, Win, Wout, b, U, Y, ubuf, cnt);
}